// development_layer_31963146617085
// MI455X (gfx1250) — compile-verified
//
#include <hip/hip_runtime.h>
#include <hip/hip_bf16.h>

// DevNet development layer for MI455X (gfx1250, wave32).
//
// Shapes: N=64, T=256, D=16, C=8, M=16.
// Strategy: one wave32 per (n,c) chain. All matmuls are FP32 WMMA
// (v_wmma_f32_16x16x4_f32 chained over K) for full precision, since the
// problem is latency-bound (~17 GFLOPs, ~1.6MB traffic) and expm + a
// 255-term ordered product would compound low-precision error.

typedef float v2f __attribute__((ext_vector_type(2)));
typedef float v8f __attribute__((ext_vector_type(8)));

#define ST 17  // padded LDS row stride (floats) to spread 64 banks

// D = A(16x16) @ B(16x16), A and B row-major in LDS with stride ST.
// A elements scaled by `ascale` on load (used for Taylor 1/k factors).
// Fragment layouts per CDNA5 ISA 7.12.2:
//   A 16x4:  lane<16 -> {A[l,k0+0],A[l,k0+1]}, lane>=16 -> {A[l-16,k0+2],A[l-16,k0+3]}
//   B 4x16:  lane     -> {B[k0+2*half+0, l%16], B[k0+2*half+1, l%16]}
__device__ __forceinline__ v8f wmma_mm16(const float* As, const float* Bs,
                                         float ascale, int half, int l16) {
  v8f c = {0.f, 0.f, 0.f, 0.f, 0.f, 0.f, 0.f, 0.f};
#pragma unroll
  for (int k0 = 0; k0 < 16; k0 += 4) {
    const int ka = k0 + 2 * half;
    v2f a, b;
    a.x = As[l16 * ST + ka] * ascale;
    a.y = As[l16 * ST + ka + 1] * ascale;
    b.x = Bs[ka * ST + l16];
    b.y = Bs[(ka + 1) * ST + l16];
    c = __builtin_amdgcn_wmma_f32_16x16x4_f32(false, a, false, b, (short)0, c,
                                              false, false);
  }
  return c;
}

// Scatter a C/D-layout accumulator back to a row-major LDS tile.
__device__ __forceinline__ void store16(float* Ds, v8f c, int half, int l16) {
#pragma unroll
  for (int r = 0; r < 8; ++r) Ds[(r + 8 * half) * ST + l16] = c[r];
}

// ---------------------------------------------------------------------------
// Kernel 1: Ap[c,d] = J @ sym(A[d,c]),  J = [[0,I],[-I,0]] (n=8).
// (J@S)[i,:] = S[i+8,:] for i<8, else -S[i-8,:].
// A layout (D,C,16,16); Ap written as (C,D,16,16) so a wave's c-slice is
// one contiguous 16KB block.
// ---------------------------------------------------------------------------
__global__ void sp_frame_kernel(const float* __restrict__ A,
                                float* __restrict__ Ap) {
  int idx = blockIdx.x * blockDim.x + threadIdx.x;  // 8*16*16*16 = 32768
  if (idx >= 8 * 16 * 256) return;
  int j = idx & 15;
  int i = (idx >> 4) & 15;
  int d = (idx >> 8) & 15;
  int c = idx >> 12;
  int si = (i < 8) ? (i + 8) : (i - 8);
  float sgn = (i < 8) ? 1.0f : -1.0f;
  const float* Adc = A + (d * 8 + c) * 256;
  float s = 0.5f * (Adc[si * 16 + j] + Adc[j * 16 + si]);
  Ap[idx] = sgn * s;
}

// ---------------------------------------------------------------------------
// Kernel 2: per-(n,c) chain.  P = prod_t expm( (1/1)*sum_d dX[t,d]*Ap[d,c] ).
// expm by scaling-and-squaring: B/64, degree-10 Taylor (Horner), 6 squarings.
// ---------------------------------------------------------------------------
__global__ __launch_bounds__(32) void devlayer_main(
    const float* __restrict__ x, const float* __restrict__ Ap,
    float* __restrict__ out) {
  __shared__ float Bm[16 * ST];
  __shared__ float Em[16 * ST];
  __shared__ float Pm[16 * ST];

  const int wg = blockIdx.x;  // 0..511
  const int n = wg >> 3;
  const int c = wg & 7;
  const int lane = threadIdx.x;
  const int half = lane >> 4;
  const int l16 = lane & 15;

  // Ap[c] resident in VGPRs: each lane keeps its 8 C/D-layout elements of
  // all 16 d-matrices (128 VGPRs/lane; occupancy is irrelevant here).
  float apreg[16][8];
  const float* apc = Ap + (c << 12);
#pragma unroll
  for (int d = 0; d < 16; ++d)
#pragma unroll
    for (int r = 0; r < 8; ++r)
      apreg[d][r] = apc[d * 256 + (r + 8 * half) * 16 + l16];

  // P = I
#pragma unroll
  for (int r = 0; r < 8; ++r) {
    int i = r + 8 * half;
    Pm[i * ST + l16] = (i == l16) ? 1.0f : 0.0f;
  }
  __builtin_amdgcn_wave_barrier();

  const float* xn = x + n * 256 * 16;

#pragma unroll 1
  for (int t = 0; t < 255; ++t) {
    const float* xr = xn + t * 16;
    __builtin_prefetch(xr + 32, 0, 1);  // next row -> global_prefetch_b8

    // AX = sum_d (x[t+1,d]-x[t,d]) * Ap[d]  (C/D-layout ownership per lane)
    float acc[8];
#pragma unroll
    for (int r = 0; r < 8; ++r) acc[r] = 0.f;
#pragma unroll
    for (int d = 0; d < 16; ++d) {
      float co = xr[16 + d] - xr[d];  // wave-uniform -> scalar loads
#pragma unroll
      for (int r = 0; r < 8; ++r) acc[r] = fmaf(co, apreg[d][r], acc[r]);
    }

    // Bm = AX/64 ; Em = I + Bm/10  (Horner seed, saves one matmul)
#pragma unroll
    for (int r = 0; r < 8; ++r) {
      int i = r + 8 * half;
      float bs = acc[r] * (1.0f / 64.0f);
      Bm[i * ST + l16] = bs;
      Em[i * ST + l16] = bs * (1.0f / 10.0f) + ((i == l16) ? 1.0f : 0.0f);
    }
    __builtin_amdgcn_wave_barrier();

    // Horner: E = I + (B/k) @ E for k = 9..1
#pragma unroll
    for (int k = 9; k >= 1; --k) {
      v8f md = wmma_mm16(Bm, Em, 1.0f / (float)k, half, l16);
#pragma unroll
      for (int r = 0; r < 8; ++r) {
        int i = r + 8 * half;
        md[r] += (i == l16) ? 1.0f : 0.0f;
      }
      __builtin_amdgcn_wave_barrier();
      store16(Em, md, half, l16);
      __builtin_amdgcn_wave_barrier();
    }

    // 6 squarings: E <- E @ E  (exp(B) = exp(B/64)^64)
#pragma unroll
    for (int s = 0; s < 6; ++s) {
      v8f md = wmma_mm16(Em, Em, 1.0f, half, l16);
      __builtin_amdgcn_wave_barrier();
      store16(Em, md, half, l16);
      __builtin_amdgcn_wave_barrier();
    }

    // chain: P <- P @ E  (dyadic_prod order: M_0 @ M_1 @ ... left-to-right)
    {
      v8f md = wmma_mm16(Pm, Em, 1.0f, half, l16);
      __builtin_amdgcn_wave_barrier();
      store16(Pm, md, half, l16);
      __builtin_amdgcn_wave_barrier();
    }
  }

  // out[n,c,i,j]
  float* o = out + ((n * 8 + c) << 8);
#pragma unroll
  for (int r = 0; r < 8; ++r) {
    int i = r + 8 * half;
    o[i * 16 + l16] = Pm[i * ST + l16];
  }
}

extern "C" void kernel_launch(void* const* d_in, const int* in_sizes, int n_in,
                              void* d_out, int out_size, void* d_ws,
                              size_t ws_size, hipStream_t stream) {
  const float* x = (const float*)d_in[0];  // (64,256,16) f32
  const float* A = (const float*)d_in[1];  // (16,8,16,16) f32
  float* out = (float*)d_out;              // (64,8,16,16) f32
  float* Ap = (float*)d_ws;                // (8,16,16,16) f32 = 128KB scratch

  sp_frame_kernel<<<128, 256, 0, stream>>>(A, Ap);
  devlayer_main<<<512, 32, 0, stream>>>(x, Ap, out);
}